// SmoothedSpheres_10754598109361
// MI455X (gfx1250) — compile-verified
//
#include <hip/hip_runtime.h>

#define N_PTS     1048576
#define N_SPHERES 128
#define K_SMOOTH  32.0f
#define CLAMP_MIN 1e-6f
#define LOG2E     1.4426950408889634f
#define LN2       0.6931471805599453f
#define KL        (K_SMOOTH * LOG2E)      /* 46.1662... */

typedef float v2f __attribute__((ext_vector_type(2)));
typedef float v8f __attribute__((ext_vector_type(8)));

// ---------------------------------------------------------------------------
// Fold (tfs + I), centers into per-sphere quadratic-form coefficients:
//   ||T p - c||^2 = p^T (T^T T) p - 2 (T^T c)·p + c·c
// Features per point: [x^2, y^2, z^2, xy, xz, yz, x, y, z, 1, 0, 0]  (K = 12)
// ---------------------------------------------------------------------------
__global__ __launch_bounds__(128) void coeff_kernel(const float* __restrict__ centers,
                                                    const float* __restrict__ tfs,
                                                    float* __restrict__ W) {
    int i = threadIdx.x;
    if (i >= N_SPHERES) return;
    float T[3][3];
#pragma unroll
    for (int r = 0; r < 3; ++r)
#pragma unroll
        for (int c = 0; c < 3; ++c)
            T[r][c] = tfs[i * 9 + r * 3 + c] + ((r == c) ? 1.0f : 0.0f);
    float cx = centers[i * 3 + 0];
    float cy = centers[i * 3 + 1];
    float cz = centers[i * 3 + 2];
    float M[3][3];
#pragma unroll
    for (int j = 0; j < 3; ++j)
#pragma unroll
        for (int k = 0; k < 3; ++k)
            M[j][k] = T[0][j] * T[0][k] + T[1][j] * T[1][k] + T[2][j] * T[2][k];
    float v0 = T[0][0] * cx + T[1][0] * cy + T[2][0] * cz;
    float v1 = T[0][1] * cx + T[1][1] * cy + T[2][1] * cz;
    float v2 = T[0][2] * cx + T[1][2] * cy + T[2][2] * cz;
    float* w = W + i * 12;
    w[0]  = M[0][0];
    w[1]  = M[1][1];
    w[2]  = M[2][2];
    w[3]  = 2.0f * M[0][1];
    w[4]  = 2.0f * M[0][2];
    w[5]  = 2.0f * M[1][2];
    w[6]  = -2.0f * v0;
    w[7]  = -2.0f * v1;
    w[8]  = -2.0f * v2;
    w[9]  = cx * cx + cy * cy + cz * cz;
    w[10] = 0.0f;
    w[11] = 0.0f;
}

// 3 chained K=4 WMMAs -> one 16x16 distance^2 tile (16 points x 16 spheres)
__device__ __forceinline__ v8f tile_wmma(const float* __restrict__ wb, int kb,
                                         v2f A0, v2f A1, v2f A2) {
    v2f B0, B1, B2;                       // B 4x16 layout mirrors A 16x4
    B0.x = wb[kb + 0]; B0.y = wb[kb + 1];
    B1.x = wb[kb + 4]; B1.y = wb[kb + 5];
    B2.x = wb[kb + 8]; B2.y = wb[kb + 9];
    v8f c = {0.f, 0.f, 0.f, 0.f, 0.f, 0.f, 0.f, 0.f};
    c = __builtin_amdgcn_wmma_f32_16x16x4_f32(false, A0, false, B0, (short)0, c, false, false);
    c = __builtin_amdgcn_wmma_f32_16x16x4_f32(false, A1, false, B1, (short)0, c, false, false);
    c = __builtin_amdgcn_wmma_f32_16x16x4_f32(false, A2, false, B2, (short)0, c, false, false);
    return c;
}

// Per-pair tail: sqrt(|q2|), exp2(fma(sqrt, -KL, rad*KL)), accumulate.
// Three phases of 8 independent chains so the scheduler can interleave
// trans-op latencies instead of emitting v_nop padding.
__device__ __forceinline__ void tile_process(const v8f& c, float radS, float* esum) {
    float t[8];
#pragma unroll
    for (int r = 0; r < 8; ++r)
        t[r] = __builtin_amdgcn_sqrtf(__builtin_fabsf(c[r]));   // |.| is free src-mod
    float e[8];
#pragma unroll
    for (int r = 0; r < 8; ++r)
        e[r] = __builtin_amdgcn_exp2f(__builtin_fmaf(t[r], -KL, radS));
#pragma unroll
    for (int r = 0; r < 8; ++r)
        esum[r] += e[r];
}

// ---------------------------------------------------------------------------
// Main kernel: 8 waves/block, 16 points per wave, all 128 spheres.
// Software-pipelined: tile tn+1's WMMAs issue before tile tn's trans tail.
// ---------------------------------------------------------------------------
__global__ __launch_bounds__(256) void sdf_kernel(const float* __restrict__ p,
                                                  const float* __restrict__ radii,
                                                  const float* __restrict__ W,
                                                  float* __restrict__ out) {
    __shared__ float sW[N_SPHERES * 12];   // 6 KB coefficient table
    __shared__ float sR[N_SPHERES];        // radii pre-scaled by K*log2(e)

    int tid = threadIdx.x;
    for (int k = tid; k < N_SPHERES * 12; k += 256) sW[k] = W[k];
    if (tid < N_SPHERES) sR[tid] = radii[tid] * KL;
    __syncthreads();

    const int wave = tid >> 5;     // wave32
    const int lane = tid & 31;
    const int half = lane >> 4;    // 0: holds K 0/1, 1: holds K 2/3 (A/B layout)
    const int lm   = lane & 15;
    const int baseM = (blockIdx.x * 8 + wave) * 16;

    // Per-point features (point m = baseM + lm; both halves mirror M=0..15)
    const int m = baseM + lm;
    const float x = p[m * 3 + 0];
    const float y = p[m * 3 + 1];
    const float z = p[m * 3 + 2];
    const float f0 = x * x, f1 = y * y, f2 = z * z;
    const float f3 = x * y, f4 = x * z, f5 = y * z;
    const float f6 = x, f7 = y, f8 = z, f9 = 1.0f, f10 = 0.0f, f11 = 0.0f;

    // A fragments per ISA 32-bit A 16x4 layout:
    //   lanes 0-15: VGPR0 = K0, VGPR1 = K1 ; lanes 16-31: VGPR0 = K2, VGPR1 = K3
    v2f A0, A1, A2;
    A0.x = half ? f2  : f0;  A0.y = half ? f3  : f1;   // K = 0..3
    A1.x = half ? f6  : f4;  A1.y = half ? f7  : f5;   // K = 4..7
    A2.x = half ? f10 : f8;  A2.y = half ? f11 : f9;   // K = 8..11

    float esum[8];
#pragma unroll
    for (int r = 0; r < 8; ++r) esum[r] = 0.0f;

    const int kb = 2 * half;   // which K pair this lane holds in A/B fragments

    // Software pipeline across the 8 sphere tiles of 16
    v8f cPrev = tile_wmma(&sW[(0 * 16 + lm) * 12], kb, A0, A1, A2);
#pragma unroll
    for (int tn = 1; tn < 8; ++tn) {
        v8f cCur = tile_wmma(&sW[(tn * 16 + lm) * 12], kb, A0, A1, A2);
        tile_process(cPrev, sR[(tn - 1) * 16 + lm], esum);
        cPrev = cCur;
    }
    tile_process(cPrev, sR[7 * 16 + lm], esum);

    // Reduce over the 16 sphere-columns held across lanes of each half-wave.
    // (xor masks 1,2,4,8 never touch bit 4, so halves stay independent.)
#pragma unroll
    for (int r = 0; r < 8; ++r) {
        float s = esum[r];
        s += __shfl_xor(s, 1, 32);
        s += __shfl_xor(s, 2, 32);
        s += __shfl_xor(s, 4, 32);
        s += __shfl_xor(s, 8, 32);
        esum[r] = s;
    }

    // C layout: VGPR r holds M=r (lanes 0-15) / M=r+8 (lanes 16-31).
    if (lm == 0) {
#pragma unroll
        for (int r = 0; r < 8; ++r) {
            float s = fmaxf(esum[r], CLAMP_MIN);
            out[baseM + half * 8 + r] = -__builtin_amdgcn_logf(s) * (LN2 / K_SMOOTH);
        }
    }
}

extern "C" void kernel_launch(void* const* d_in, const int* in_sizes, int n_in,
                              void* d_out, int out_size, void* d_ws, size_t ws_size,
                              hipStream_t stream) {
    const float* p       = (const float*)d_in[0];
    const float* centers = (const float*)d_in[1];
    const float* radii   = (const float*)d_in[2];
    const float* tfs     = (const float*)d_in[3];
    float* W   = (float*)d_ws;     // 128 x 12 coefficient table (6 KB)
    float* out = (float*)d_out;

    coeff_kernel<<<1, 128, 0, stream>>>(centers, tfs, W);
    sdf_kernel<<<N_PTS / 128, 256, 0, stream>>>(p, radii, W, out);
}